// DKVMN_5179730559460
// MI455X (gfx1250) — compile-verified
//
#include <hip/hip_runtime.h>
#include <hip/hip_bf16.h>
#include <math.h>

typedef __attribute__((ext_vector_type(2))) float v2f;
typedef __attribute__((ext_vector_type(8))) float v8f;

#define BSZ   64
#define LSEQ  200
#define DIM   128
#define MSLOT 50
#define NQ    10000
#define BL    (BSZ * LSEQ)   // 12800

// Full-precision f32 WMMA: D = A(16x4) * B(4x16) + C(16x16)
__device__ __forceinline__ v8f wmma4(v2f a, v2f b, v8f c) {
    return __builtin_amdgcn_wmma_f32_16x16x4_f32(
        /*neg_a=*/false, a, /*neg_b=*/false, b,
        /*c_mod=*/(short)0, c, /*reuse_a=*/false, /*reuse_b=*/false);
}

__device__ __forceinline__ float sigmoidf_(float x) { return 1.0f / (1.0f + __expf(-x)); }

// ---------------------------------------------------------------------------
// Kernel 1: gather embedding rows.  grid=BL blocks, block=128 threads
// ---------------------------------------------------------------------------
__global__ void k_gather(const int* __restrict__ q, const int* __restrict__ r,
                         const float* __restrict__ k_emb, const float* __restrict__ v_emb,
                         float* __restrict__ kbuf, float* __restrict__ vbuf) {
    int i   = blockIdx.x;
    int tid = threadIdx.x;
    int qi  = q[i];
    int xi  = qi + NQ * r[i];
    kbuf[(size_t)i * DIM + tid] = k_emb[(size_t)qi * DIM + tid];
    vbuf[(size_t)i * DIM + tid] = v_emb[(size_t)xi * DIM + tid];
}

// ---------------------------------------------------------------------------
// Kernel 2: ea = act(v @ [e_W | a_W]^T + bias)  -> [BL, 256]
// one wave per 16x16 output tile; 12800 tiles; block = 256 (8 waves), grid=1600
// ---------------------------------------------------------------------------
__global__ void k_ea_gemm(const float* __restrict__ vbuf,
                          const float* __restrict__ e_W, const float* __restrict__ a_W,
                          const float* __restrict__ e_b, const float* __restrict__ a_b,
                          float* __restrict__ ea) {
    int wave  = threadIdx.x >> 5;
    int lane  = threadIdx.x & 31;
    int tile  = blockIdx.x * 8 + wave;     // 0..12799
    int rt    = tile >> 4;                 // row tile 0..799
    int ct    = tile & 15;                 // col tile 0..15
    int R     = rt * 16;
    int ln    = lane & 15;
    int khalf = (lane >> 4) << 1;          // 0 or 2
    int n_g   = ct * 16 + ln;              // global output col 0..255

    const float* Bbase = (n_g < DIM) ? (e_W + (size_t)n_g * DIM)
                                     : (a_W + (size_t)(n_g - DIM) * DIM);
    const float* Abase = vbuf + (size_t)(R + ln) * DIM;

    v8f acc = {};
    for (int k0 = 0; k0 < DIM; k0 += 4) {
        v2f a = *(const v2f*)(Abase + k0 + khalf);
        v2f b = *(const v2f*)(Bbase + k0 + khalf);
        acc = wmma4(a, b, acc);
    }

    float bias = (n_g < DIM) ? e_b[n_g] : a_b[n_g - DIM];
    int mbase = (lane < 16) ? 0 : 8;
#pragma unroll
    for (int i2 = 0; i2 < 8; ++i2) {
        int m   = mbase + i2;
        float x = acc[i2] + bias;
        float y = (n_g < DIM) ? sigmoidf_(x) : tanhf(x);
        ea[(size_t)(R + m) * 256 + n_g] = y;
    }
}

// ---------------------------------------------------------------------------
// Kernel 3: logits = k @ Mk^T (cols padded 50->64), softmax -> w [BL, 64]
// one wave per 16-row tile; grid = 800, block = 32
// ---------------------------------------------------------------------------
__global__ void k_w_gemm(const float* __restrict__ kbuf, const float* __restrict__ Mk,
                         float* __restrict__ wbuf) {
    __shared__ float lg[16 * 64];
    int lane  = threadIdx.x & 31;
    int R     = blockIdx.x * 16;
    int ln    = lane & 15;
    int khalf = (lane >> 4) << 1;
    const float* Abase = kbuf + (size_t)(R + ln) * DIM;

    v8f acc[4] = {};
    for (int k0 = 0; k0 < DIM; k0 += 4) {
        v2f a = *(const v2f*)(Abase + k0 + khalf);
#pragma unroll
        for (int c = 0; c < 4; ++c) {
            int n = c * 16 + ln;
            v2f b = {0.0f, 0.0f};
            if (n < MSLOT) b = *(const v2f*)(Mk + (size_t)n * DIM + k0 + khalf);
            acc[c] = wmma4(a, b, acc[c]);
        }
    }

    int mbase = (lane < 16) ? 0 : 8;
#pragma unroll
    for (int c = 0; c < 4; ++c)
#pragma unroll
        for (int i2 = 0; i2 < 8; ++i2)
            lg[(mbase + i2) * 64 + c * 16 + ln] = acc[c][i2];
    __syncthreads();

    if (lane < 16) {
        float mx = -1e30f;
        for (int m = 0; m < MSLOT; ++m) mx = fmaxf(mx, lg[lane * 64 + m]);
        float s = 0.0f;
        for (int m = 0; m < MSLOT; ++m) {
            float e = __expf(lg[lane * 64 + m] - mx);
            lg[lane * 64 + m] = e;
            s += e;
        }
        float inv = 1.0f / s;
        for (int m = 0; m < MSLOT; ++m)
            wbuf[(size_t)(R + lane) * 64 + m] = lg[lane * 64 + m] * inv;
    }
}

// ---------------------------------------------------------------------------
// Kernel 4: sequential memory scan.  grid = B (64), block = 256
// Mv state [50,128] lives in LDS; streams all 201 states to outMv with
// non-temporal stores (329 MB write-once stream > 192 MB L2 -> keep L2 for
// the reused ea/w/read/k workspace buffers).
// ---------------------------------------------------------------------------
__global__ void k_scan(const float* __restrict__ Mv0, const float* __restrict__ wbuf,
                       const float* __restrict__ ea, float* __restrict__ readbuf,
                       float* __restrict__ outMv) {
    __shared__ float sMv[MSLOT * DIM];   // 25.6 KB
    __shared__ float sw[64];
    __shared__ float sea[256];           // [0:128)=e, [128:256)=a

    int b   = blockIdx.x;
    int tid = threadIdx.x;
    float* myOut = outMv + (size_t)b * (LSEQ + 1) * MSLOT * DIM;

    for (int i = tid; i < MSLOT * DIM; i += 256) {
        float v = Mv0[i];
        sMv[i]  = v;
        __builtin_nontemporal_store(v, &myOut[i]);   // Mv state at t=0 (init)
    }
    __syncthreads();

    for (int t = 0; t < LSEQ; ++t) {
        int idx = b * LSEQ + t;
        if (tid < 64) sw[tid] = wbuf[(size_t)idx * 64 + tid];
        sea[tid] = ea[(size_t)idx * 256 + tid];
        __syncthreads();

        // read_t uses state BEFORE this step's update
        if (tid < DIM) {
            float acc = 0.0f;
            for (int m = 0; m < MSLOT; ++m) acc += sw[m] * sMv[m * DIM + tid];
            readbuf[(size_t)idx * DIM + tid] = acc;
        }
        __syncthreads();

        float* outT = myOut + (size_t)(t + 1) * MSLOT * DIM;
        for (int i = tid; i < MSLOT * DIM; i += 256) {
            int m = i >> 7;
            int d = i & 127;
            float wm = sw[m];
            float nv = sMv[i] * (1.0f - wm * sea[d]) + wm * sea[DIM + d];
            sMv[i] = nv;
            __builtin_nontemporal_store(nv, &outT[i]);  // dominant 329 MB stream
        }
        __syncthreads();
    }
}

// ---------------------------------------------------------------------------
// Kernel 5: f = tanh([read,k] @ f_W^T + f_b); p = sigmoid(f . p_W + p_b)
// one wave per 16 rows x all 8 col tiles; grid = 800, block = 32
// ---------------------------------------------------------------------------
__global__ void k_f_gemm(const float* __restrict__ readbuf, const float* __restrict__ kbuf,
                         const float* __restrict__ f_W, const float* __restrict__ f_b,
                         const float* __restrict__ p_W, const float* __restrict__ p_b,
                         float* __restrict__ pout) {
    __shared__ float sf[16 * DIM];
    int lane  = threadIdx.x & 31;
    int R     = blockIdx.x * 16;
    int ln    = lane & 15;
    int khalf = (lane >> 4) << 1;

    v8f acc[8] = {};
    for (int k0 = 0; k0 < 2 * DIM; k0 += 4) {
        int kk = k0 + khalf;             // step 4, 128-aligned regions: float2 never straddles
        const float* Ap = (kk < DIM) ? (readbuf + (size_t)(R + ln) * DIM + kk)
                                     : (kbuf + (size_t)(R + ln) * DIM + (kk - DIM));
        v2f a = *(const v2f*)Ap;
#pragma unroll
        for (int c = 0; c < 8; ++c) {
            int n = c * 16 + ln;
            v2f bb = *(const v2f*)(f_W + (size_t)n * 256 + kk);
            acc[c] = wmma4(a, bb, acc[c]);
        }
    }

    int mbase = (lane < 16) ? 0 : 8;
#pragma unroll
    for (int c = 0; c < 8; ++c) {
        int n = c * 16 + ln;
        float bias = f_b[n];
#pragma unroll
        for (int i2 = 0; i2 < 8; ++i2)
            sf[(mbase + i2) * DIM + n] = tanhf(acc[c][i2] + bias);
    }
    __syncthreads();

    if (lane < 16) {
        float acc2 = p_b[0];
        for (int d = 0; d < DIM; ++d) acc2 += sf[lane * DIM + d] * p_W[d];
        pout[R + lane] = sigmoidf_(acc2);
    }
}

// ---------------------------------------------------------------------------
extern "C" void kernel_launch(void* const* d_in, const int* in_sizes, int n_in,
                              void* d_out, int out_size, void* d_ws, size_t ws_size,
                              hipStream_t stream) {
    const int*   q     = (const int*)d_in[0];
    const int*   r     = (const int*)d_in[1];
    const float* k_emb = (const float*)d_in[2];
    const float* v_emb = (const float*)d_in[3];
    const float* Mk    = (const float*)d_in[4];
    const float* Mv0   = (const float*)d_in[5];
    const float* f_W   = (const float*)d_in[6];
    const float* f_b   = (const float*)d_in[7];
    const float* p_W   = (const float*)d_in[8];
    const float* p_b   = (const float*)d_in[9];
    const float* e_W   = (const float*)d_in[10];
    const float* e_b   = (const float*)d_in[11];
    const float* a_W   = (const float*)d_in[12];
    const float* a_b   = (const float*)d_in[13];

    float* ws      = (float*)d_ws;
    float* kbuf    = ws;                            // BL*128
    float* vbuf    = kbuf + (size_t)BL * DIM;       // BL*128 (reused as readbuf)
    float* ea      = vbuf + (size_t)BL * DIM;       // BL*256
    float* wbuf    = ea + (size_t)BL * 256;         // BL*64
    float* readbuf = vbuf;                          // safe: vbuf consumed by k_ea_gemm

    float* pout  = (float*)d_out;                   // [BL]
    float* outMv = pout + BL;                       // [B, L+1, M, D]

    hipLaunchKernelGGL(k_gather, dim3(BL), dim3(DIM), 0, stream,
                       q, r, k_emb, v_emb, kbuf, vbuf);
    hipLaunchKernelGGL(k_ea_gemm, dim3(1600), dim3(256), 0, stream,
                       vbuf, e_W, a_W, e_b, a_b, ea);      // 12800 tiles / 8 waves
    hipLaunchKernelGGL(k_w_gemm, dim3(BL / 16), dim3(32), 0, stream,
                       kbuf, Mk, wbuf);
    hipLaunchKernelGGL(k_scan, dim3(BSZ), dim3(256), 0, stream,
                       Mv0, wbuf, ea, readbuf, outMv);
    hipLaunchKernelGGL(k_f_gemm, dim3(BL / 16), dim3(32), 0, stream,
                       readbuf, kbuf, f_W, f_b, p_W, p_b, pout);
}